// WaveletPooling_26018911879865
// MI455X (gfx1250) — compile-verified
//
#include <hip/hip_runtime.h>
#include <cstdint>

// out = cA1 of the Haar DWT: out[n,c,i,j] = 0.5*(x[2i,2j]+x[2i,2j+1]+x[2i+1,2j]+x[2i+1,2j+1])
// (the reference's second analysis + synthesis is an exact round-trip identity:
//  x00 = 0.5*(cA+cH+cV+cD) = 0.25*4*x00).
//
// Shapes fixed by the reference:
//   x:   (16,128,256,256) fp32 -> 2048 images of 256x256
//   out: (16,128,128,128) fp32 -> 2048 images of 128x128
//
// Roofline: ~671 MB traffic, ~168 MFLOP -> pure HBM streaming (~29 us floor
// at 23.3 TB/s). WMMA inapplicable; the CDNA5 path that matters is the
// ASYNCcnt pipeline: global_load_async_to_lds_b128 double-buffered against
// ds_load_b128 consumption, with s_wait_asynccnt for overlap.

typedef float v2f __attribute__((ext_vector_type(2)));
typedef float v4f __attribute__((ext_vector_type(4)));

#define TPB 256u
#define IPT 8u
// total items = 16*128*128*64 = 2^24 float2 outputs
// grid = 2^24 / (TPB*IPT) = 8192 blocks; thread stride = 2^21 items.
// 2^21 is a multiple of 8192 (items/image), so between a thread's successive
// items only img changes (+256)  =>  constant address steps:
#define GSTEP_BYTES  (256ull * 65536ull * 4ull)   // +64 MiB input per item
#define OSTEP_FLOATS (2097152u * 2u)              // +2^22 output floats per item

__global__ __launch_bounds__(TPB)
void haar_cA1_async_kernel(const float* __restrict__ x, float* __restrict__ out)
{
    // [buffer][input-row (2i / 2i+1)][thread * 4 floats]
    __shared__ __align__(16) float smem[2][2][TPB * 4];

    const uint32_t t    = threadIdx.x;
    const uint32_t idx0 = blockIdx.x * TPB + t;

    // addrspace(3) -> generic; low 32 bits are the wave-relative LDS byte
    // offset — what VDST of the async load
    // expects.
    const uint32_t lds0 = (uint32_t)(uintptr_t)(&smem[0][0][0]) + t * 16u;

    // One-time mapping for item k=0; later items are constant-stride.
    const uint32_t col2 = idx0 & 63u;          // 64 float2 per output row
    const uint32_t row  = (idx0 >> 6) & 127u;  // 128 output rows / image
    const uint32_t img  = idx0 >> 13;          // 8192 items per image
    uint64_t ga = (uint64_t)(uintptr_t)
        (x + (((size_t)img << 16) + ((size_t)row << 9) + ((size_t)col2 << 2)));
    float* op = out + ((size_t)idx0 << 1);

    auto issue = [&](uint64_t a0, uint32_t buf) {
        const uint32_t l0 = lds0 + buf * (2u * TPB * 16u);
        const uint32_t l1 = l0 + TPB * 16u;
        const uint64_t a1 = a0 + 256ull * sizeof(float);     // input row 2i+1
        asm volatile("global_load_async_to_lds_b128 %0, %1, off"
                     :: "v"(l0), "v"(a0) : "memory");
        asm volatile("global_load_async_to_lds_b128 %0, %1, off"
                     :: "v"(l1), "v"(a1) : "memory");
    };

    issue(ga, 0);

#pragma unroll
    for (uint32_t k = 0; k < IPT; ++k) {
        const uint32_t buf = k & 1u;
        if (k + 1u < IPT) {
            issue(ga + GSTEP_BYTES, buf ^ 1u);                 // prefetch next
            asm volatile("s_wait_asynccnt 0x2" ::: "memory");  // current done
        } else {
            asm volatile("s_wait_asynccnt 0x0" ::: "memory");
        }

        const v4f a = *(const v4f*)&smem[buf][0][t * 4];  // x[2i,   4c..4c+3]
        const v4f b = *(const v4f*)&smem[buf][1][t * 4];  // x[2i+1, 4c..4c+3]

        v2f o;
        o.x = 0.5f * ((a.x + a.y) + (b.x + b.y));
        o.y = 0.5f * ((a.z + a.w) + (b.z + b.w));
        __builtin_nontemporal_store(o, (v2f*)op);

        ga += GSTEP_BYTES;       // next item: img += 256, row/col unchanged
        op += OSTEP_FLOATS;
    }
}

extern "C" void kernel_launch(void* const* d_in, const int* in_sizes, int n_in,
                              void* d_out, int out_size, void* d_ws, size_t ws_size,
                              hipStream_t stream) {
    const float* x   = (const float*)d_in[0];
    float*       out = (float*)d_out;

    constexpr uint32_t ITEMS = 16u * 128u * 128u * 64u;  // 2^24 float2 outputs
    const dim3 grid(ITEMS / (TPB * IPT));                // 8192 blocks
    const dim3 block(TPB);                               // 8 wave32s

    haar_cA1_async_kernel<<<grid, block, 0, stream>>>(x, out);
}